// PyramidROIAlign_58841051955937
// MI455X (gfx1250) — compile-verified
//
#include <hip/hip_runtime.h>

// PyramidROIAlign for MI455X (gfx1250, wave32).
// out[roi, 7,7, 256] = bilinear crop_and_resize on the pyramid level chosen per ROI.
// Formulated per 16-point M-tile as D(16x16) += A(16x4) x B(4x16) f32 WMMA,
// K-looped over the 64 gathered source pixel-vectors staged in LDS.

typedef float v2f __attribute__((ext_vector_type(2)));
typedef float v8f __attribute__((ext_vector_type(8)));

#define CH       256
#define NPTS     49      // 7x7 pool
#define GROWS    64      // 16 points * 4 corners per M-tile
#define GSTRIDE  272     // padded LDS row stride (floats): 2*272 % 64 == 32 -> halves hit disjoint banks

__global__ __launch_bounds__(256) void pyramid_roi_align_wmma(
    const float* __restrict__ rois,   // [N,5] (b, y1, x1, y2, x2)
    const float* __restrict__ fm2,    // [2,256,256,256]
    const float* __restrict__ fm3,    // [2,128,128,256]
    const float* __restrict__ fm4,    // [2, 64, 64,256]
    const float* __restrict__ fm5,    // [2, 32, 32,256]
    const float* __restrict__ meta,   // [2,11]
    float* __restrict__ out,          // [N,7,7,256]
    int nroi)
{
    __shared__ float G[GROWS * GSTRIDE];   // gathered pixel vectors (B-matrix source)
    __shared__ int   offs[GROWS];          // element offsets into selected fm

    const int roi  = blockIdx.x;
    if (roi >= nroi) return;
    const int tid  = threadIdx.x;
    const int lane = tid & 31;
    const int wave = tid >> 5;
    const int half = lane >> 4;   // 0: K={0,1} (tl,tr)   1: K={2,3} (bl,br)
    const int n    = lane & 15;   // A row (point-in-tile) / B,D column (channel)

    // ---------------- per-ROI uniform setup ----------------
    const float y1 = rois[roi * 5 + 1];
    const float x1 = rois[roi * 5 + 2];
    const float y2 = rois[roi * 5 + 3];
    const float x2 = rois[roi * 5 + 4];
    const int   b  = (int)rois[roi * 5 + 0];

    const float areas = (float)((int)meta[6]) * (float)((int)meta[7]);
    const float bh = y2 - y1, bw = x2 - x1;
    float lvlf = log2f(sqrtf(bh * bw) / (224.0f / sqrtf(areas)));
    lvlf = 4.0f + rintf(lvlf);                       // RNE matches jnp.round
    lvlf = fminf(fmaxf(lvlf, 2.0f), 5.0f);
    const int level = (int)lvlf;

    const float* __restrict__ fm =
        (level == 2) ? fm2 : (level == 3) ? fm3 : (level == 4) ? fm4 : fm5;
    const int H = 256 >> (level - 2);
    const int W = H;

    const float oy = y1 * (float)(H - 1);
    const float ox = x1 * (float)(W - 1);
    const float sy = bh * (float)(H - 1) * (1.0f / 6.0f);
    const float sx = bw * (float)(W - 1) * (1.0f / 6.0f);
    const float ymax = (float)(H - 1), xmax = (float)(W - 1);

    for (int mt = 0; mt < 4; ++mt) {
        const int p0 = mt * 16;

        // ---- source offsets for this M-tile (64 gathered pixel vectors) ----
        if (tid < GROWS) {
            const int pl = tid >> 2, j = tid & 3;        // j: 0=tl 1=tr 2=bl 3=br
            int p = p0 + pl; if (p > 48) p = 48;         // pad rows reuse point 48 (never stored)
            const int   iy  = p / 7, ix = p % 7;
            const float iny = oy + (float)iy * sy;
            const float inx = ox + (float)ix * sx;
            const int   yf  = (int)floorf(iny);
            const int   xf  = (int)floorf(inx);
            const int   ya  = min(max(yf,     0), H - 1);
            const int   yb  = min(max(yf + 1, 0), H - 1);
            const int   xa  = min(max(xf,     0), W - 1);
            const int   xb  = min(max(xf + 1, 0), W - 1);
            const int   yy  = (j & 2) ? yb : ya;
            const int   xx  = (j & 1) ? xb : xa;
            offs[tid] = ((b * H + yy) * W + xx) * CH;
        }
        __syncthreads();

        // ---- gather 64 x 1KB pixel vectors into LDS (coalesced B128 loads) ----
        #pragma unroll
        for (int i = 0; i < 16; ++i) {
            const int flat  = i * 256 + tid;
            const int row   = flat >> 6;                 // 64 float4 chunks per row
            const int chunk = flat & 63;
            const float4 v  = *(const float4*)(fm + offs[row] + chunk * 4);
            *(float4*)(&G[row * GSTRIDE + chunk * 4]) = v;
        }
        __syncthreads();

        // ---- per-lane bilinear weights for A (lane n = point p0+n) ----
        {
            int p = p0 + n; if (p > 48) p = 48;
            const int   iy  = p / 7, ix = p % 7;
            const float iny = oy + (float)iy * sy;
            const float inx = ox + (float)ix * sx;
            const float ly  = iny - floorf(iny);
            const float lx  = inx - floorf(inx);
            const float vm  = (iny >= 0.f && iny <= ymax && inx >= 0.f && inx <= xmax) ? 1.f : 0.f;
            const float wy  = half ? ly : (1.f - ly);
            const float wA  = vm * wy * (1.f - lx);      // K = 2*half + 0  (x0 tap)
            const float wB  = vm * wy * lx;              // K = 2*half + 1  (x1 tap)

            // ---- WMMA: each wave computes channel tiles {wave, wave+8} ----
            for (int cc = 0; cc < 2; ++cc) {
                const int ctile = wave + cc * 8;
                const float* Bbase = &G[(2 * half) * GSTRIDE + ctile * 16 + n];
                v8f acc = {};
                #pragma unroll
                for (int k = 0; k < 16; ++k) {           // K-steps over 64 gathered rows
                    const float sel = (k == n) ? 1.f : 0.f;  // block-diagonal A
                    v2f a, bv;
                    a.x  = sel * wA;
                    a.y  = sel * wB;
                    bv.x = Bbase[(4 * k    ) * GSTRIDE];
                    bv.y = Bbase[(4 * k + 1) * GSTRIDE];
                    acc = __builtin_amdgcn_wmma_f32_16x16x4_f32(
                              false, a, false, bv, (short)0, acc, false, false);
                }
                // ---- store D: VGPR r -> rows {r, r+8}, lane n = channel ----
                const size_t obase = (size_t)roi * (NPTS * CH) + (size_t)ctile * 16 + n;
                #pragma unroll
                for (int r = 0; r < 8; ++r) {
                    const int p2 = p0 + r + 8 * half;
                    if (p2 < NPTS) out[obase + (size_t)p2 * CH] = acc[r];
                }
            }
        }
        __syncthreads();
    }
}

extern "C" void kernel_launch(void* const* d_in, const int* in_sizes, int n_in,
                              void* d_out, int out_size, void* d_ws, size_t ws_size,
                              hipStream_t stream) {
    const float* rois = (const float*)d_in[0];
    const float* fm2  = (const float*)d_in[1];
    const float* fm3  = (const float*)d_in[2];
    const float* fm4  = (const float*)d_in[3];
    const float* fm5  = (const float*)d_in[4];
    const float* meta = (const float*)d_in[5];
    float* out = (float*)d_out;
    const int nroi = in_sizes[0] / 5;

    pyramid_roi_align_wmma<<<nroi, 256, 0, stream>>>(
        rois, fm2, fm3, fm4, fm5, meta, out, nroi);
}